// SingleConnectedLayer_22789096473429
// MI455X (gfx1250) — compile-verified
//
#include <hip/hip_runtime.h>

// y[b,f] = W0[f] + W1[f] * x[b,f]   (B=4096, F=8192, fp32)
//
// Roofline: 256 MiB streamed @ 23.3 TB/s => ~11.5 us floor; 64 MFLOP is noise.
// Bandwidth-bound -> B128 NT loads/stores, register-cached W0/W1, and a
// completely branch-free hot kernel (divisibility checked on the host).

typedef float v4f __attribute__((ext_vector_type(4)));

#define TPB  256   // 8 wave32s per block
#define ROWS 8     // rows per thread: W0/W1 register reuse, 8-deep load clause

// ---- Fast kernel: requires F % (4*TPB) == 0 and B % ROWS == 0. ----------
// No guards, no branches: one load clause, one FMA burst, one store clause.
__global__ __launch_bounds__(TPB)
void scl_fast_kernel(const float* __restrict__ x,
                     const float* __restrict__ W0,
                     const float* __restrict__ W1,
                     float* __restrict__ out,
                     int f4count)            // = F/4
{
    const int f4 = blockIdx.x * TPB + threadIdx.x;

    // Per-column parameters: regular temporal (64 KiB total, grid-wide reuse
    // -> resident in WGP$/L2), held in VGPRs across ROWS rows.
    const v4f w0 = *reinterpret_cast<const v4f*>(W0 + 4 * (size_t)f4);
    const v4f w1 = *reinterpret_cast<const v4f*>(W1 + 4 * (size_t)f4);

    const long base = (long)blockIdx.y * ROWS * f4count + f4;
    const v4f* __restrict__ xp = reinterpret_cast<const v4f*>(x)  + base;
    v4f*       __restrict__ op = reinterpret_cast<v4f*>(out)      + base;

    // Warm GL2 for the Y-adjacent block's tile (gfx1250 global_prefetch_b8:
    // no VGPR return, no counter, speculative).
    __builtin_prefetch((const void*)(xp + (long)ROWS * f4count), 0, 0);

    v4f xv[ROWS];
#pragma unroll
    for (int r = 0; r < ROWS; ++r)          // 8-deep NT b128 load clause
        xv[r] = __builtin_nontemporal_load(xp + (long)r * f4count);
#pragma unroll
    for (int r = 0; r < ROWS; ++r) {        // 32x v_fma_f32 + NT store clause
        v4f y = w0 + xv[r] * w1;
        __builtin_nontemporal_store(y, op + (long)r * f4count);
    }
}

// ---- Generic fallback (any B, F with F % 4 == 0): never runs for the
// reference shapes; kept for correctness on arbitrary inputs. -------------
__global__ __launch_bounds__(TPB)
void scl_generic_kernel(const float* __restrict__ x,
                        const float* __restrict__ W0,
                        const float* __restrict__ W1,
                        float* __restrict__ out,
                        long n, int F)
{
    long i = (long)blockIdx.x * TPB + threadIdx.x;      // element index
    const long stride = (long)gridDim.x * TPB;
    for (; i < n; i += stride) {
        const int f = (int)(i % F);
        out[i] = fmaf(W1[f], x[i], W0[f]);
    }
}

extern "C" void kernel_launch(void* const* d_in, const int* in_sizes, int n_in,
                              void* d_out, int out_size, void* d_ws, size_t ws_size,
                              hipStream_t stream)
{
    const float* x  = (const float*)d_in[0];   // (B, F) fp32
    const float* W0 = (const float*)d_in[1];   // (F,)  fp32
    const float* W1 = (const float*)d_in[2];   // (F,)  fp32
    float* out = (float*)d_out;                // (B, F) fp32

    const int F = in_sizes[1];                 // 8192
    const int B = in_sizes[0] / F;             // 4096

    if ((F % (4 * TPB) == 0) && (B % ROWS == 0)) {
        const int f4count = F >> 2;            // 2048
        dim3 grid(f4count / TPB,               // 8
                  B / ROWS);                   // 512  -> 4096 blocks, 32K waves
        scl_fast_kernel<<<grid, TPB, 0, stream>>>(x, W0, W1, out, f4count);
    } else {
        const long n = (long)B * F;
        int blocks = (int)((n + TPB - 1) / TPB);
        if (blocks > 65535 * 16) blocks = 65535 * 16;
        scl_generic_kernel<<<blocks, TPB, 0, stream>>>(x, W0, W1, out, n, F);
    }
}